// Synthesis_26980984553733
// MI455X (gfx1250) — compile-verified
//
#include <hip/hip_runtime.h>

// ---------------------------------------------------------------------------
// StyleGAN2-ish synthesis network for MI455X (gfx1250).
//  - style linears + 3x3 modulated convs run as implicit GEMM on
//    v_wmma_f32_16x16x4_f32 (M tiles of pixels, N=16 cout, K stepped by 4)
//  - conv wave processes 4 M-tiles (64 pixels) with 4 accumulators so each
//    weight (B) fragment is reused 4x and WMMAs interleave across accs
//  - modulation folded into input, demodulation folded into conv epilogue
//  - upsample/blur/rgb are bandwidth-bound grid-stride kernels
// ---------------------------------------------------------------------------

typedef float v2f __attribute__((ext_vector_type(2)));
typedef float v8f __attribute__((ext_vector_type(8)));

#define LRELU 0.2f
#define DEMOD_EPS 1e-8f

static __device__ __forceinline__ int iclamp(int v, int lo, int hi) {
  return v < lo ? lo : (v > hi ? hi : v);
}

// ---------------------------------------------------------------------------
// fill
__global__ void k_fill(float* __restrict__ p, long n, float v) {
  long i = (long)blockIdx.x * blockDim.x + threadIdx.x;
  long stride = (long)gridDim.x * blockDim.x;
  for (; i < n; i += stride) p[i] = v;
}

// ---------------------------------------------------------------------------
// S[b,o] = sum_k styles[b,k] * W[o,k] + bias[o] + 1   (b=0..15, K=512)
// one wave per 16-wide tile of o. WMMA f32 16x16x4.
__global__ __launch_bounds__(32) void k_style_gemm(
    const float* __restrict__ s, const float* __restrict__ W,
    const float* __restrict__ bias, float* __restrict__ out, int outdim) {
  const int t = threadIdx.x;
  const int half = t >> 4;      // 0: K=k+0..1, 1: K=k+2..3
  const int lid = t & 15;       // A: M row  /  B: N col
  const int n0 = blockIdx.x * 16;
  const int nB = iclamp(n0 + lid, 0, outdim - 1);
  const float wmsk = (n0 + lid) < outdim ? 1.0f : 0.0f;

  const float* arow = s + (size_t)lid * 512 + half * 2;
  const float* brow = W + (size_t)nB * 512 + half * 2;

  v8f acc = {};
  for (int k = 0; k < 512; k += 4) {
    v2f a = *(const v2f*)(arow + k);
    v2f b = *(const v2f*)(brow + k);
    b *= wmsk;
    acc = __builtin_amdgcn_wmma_f32_16x16x4_f32(
        false, a, false, b, (short)0, acc, false, false);
  }
  const int oc = n0 + lid;   // C/D: N = lane&15
  if (oc < outdim) {
    const float bb = bias[oc] + 1.0f;
#pragma unroll
    for (int v = 0; v < 8; ++v) {
      const int m = v + half * 8;          // batch index
      out[(size_t)m * outdim + oc] = acc[v] + bb;
    }
  }
}

// ---------------------------------------------------------------------------
// Wsq[o,i] = sum_{r<9} w[o,i,r]^2
__global__ void k_wsq(const float* __restrict__ w, float* __restrict__ wsq, long n) {
  long i = (long)blockIdx.x * blockDim.x + threadIdx.x;
  long stride = (long)gridDim.x * blockDim.x;
  for (; i < n; i += stride) {
    float s = 0.f;
#pragma unroll
    for (int r = 0; r < 9; ++r) { float v = w[i * 9 + r]; s += v * v; }
    wsq[i] = s;
  }
}

// d[b,o] = rsqrt( sum_i sp[b,i]^2 * Wsq[o,i] + eps )
__global__ void k_demod(const float* __restrict__ sp, const float* __restrict__ wsq,
                        float* __restrict__ d, int cin, int cout) {
  int idx = blockIdx.x * blockDim.x + threadIdx.x;
  if (idx >= 16 * cout) return;
  int b = idx / cout, o = idx % cout;
  const float* sb = sp + (size_t)b * cin;
  const float* wq = wsq + (size_t)o * cin;
  float sum = 0.f;
  for (int i = 0; i < cin; ++i) { float sv = sb[i]; sum += sv * sv * wq[i]; }
  d[idx] = rsqrtf(sum + DEMOD_EPS);
}

// x[b,ci,p] *= sp[b,ci]
__global__ void k_modulate(float* __restrict__ x, const float* __restrict__ sp,
                           int cin, long hw) {
  long n = 16L * cin * hw;
  long i = (long)blockIdx.x * blockDim.x + threadIdx.x;
  long stride = (long)gridDim.x * blockDim.x;
  for (; i < n; i += stride) {
    long ch = i / hw;
    x[i] *= sp[ch];          // ch = b*cin + ci exactly
  }
}

// ---------------------------------------------------------------------------
// 3x3 same-pad conv, implicit GEMM on WMMA f32 16x16x4.
// Each wave handles 64 pixels (4 M-tiles, 4 accumulators) x 16 couts.
// grid = (ceil(HW/64), ceil(cout/16), 16 batches), 1 wave/block.
// y[b,o,p] = leaky( d[b,o] * sum_{ci,r} W[o,ci,r] * x[b,ci,p+r] )
#define MT 4
__global__ __launch_bounds__(32) void k_conv3x3(
    const float* __restrict__ x, const float* __restrict__ w,
    const float* __restrict__ dmod, float* __restrict__ y,
    int cin, int cout, int H, int Wd) {
  const int HW = H * Wd;
  const int t = threadIdx.x;
  const int half = t >> 4;
  const int lid = t & 15;
  const int m0 = blockIdx.x * (16 * MT);
  const int n0 = blockIdx.y * 16;
  const int b = blockIdx.z;

  // B-side output channel for this lane (N = lid)
  const int nB = iclamp(n0 + lid, 0, cout - 1);
  const float wmsk = (n0 + lid) < cout ? 1.0f : 0.0f;

  const float* xb = x + (size_t)b * cin * HW;
  const float* wb = w + (size_t)nB * cin * 9;
  __builtin_prefetch(wb, 0, 3);   // global_prefetch_b8: pull weight row toward WGP

  // A-side pixels for this lane, one per M-tile (M = lid within tile)
  int ph[MT], pw[MT];
  bool pv[MT];
#pragma unroll
  for (int tt = 0; tt < MT; ++tt) {
    const int pA = m0 + tt * 16 + lid;
    pv[tt] = pA < HW;
    const int pc = pv[tt] ? pA : HW - 1;
    ph[tt] = pc / Wd;
    pw[tt] = pc % Wd;
  }

  v8f acc[MT];
#pragma unroll
  for (int tt = 0; tt < MT; ++tt) acc[tt] = (v8f){};

  for (int r = 0; r < 9; ++r) {
    const int dy = r / 3 - 1, dx = r % 3 - 1;
    const float* xp[MT];
    float amsk[MT];
#pragma unroll
    for (int tt = 0; tt < MT; ++tt) {
      const int ih = ph[tt] + dy, iw = pw[tt] + dx;
      const bool inb =
          pv[tt] && (unsigned)ih < (unsigned)H && (unsigned)iw < (unsigned)Wd;
      amsk[tt] = inb ? 1.0f : 0.0f;
      const int ihc = iclamp(ih, 0, H - 1), iwc = iclamp(iw, 0, Wd - 1);
      xp[tt] = xb + (size_t)ihc * Wd + iwc;   // + ci*HW
    }
    const float* wp = wb + r;                 // + ci*9

    for (int ci = half * 2; ci < cin; ci += 4) {
      v2f bm;
      bm.x = wp[(size_t)ci * 9] * wmsk;
      bm.y = wp[(size_t)(ci + 1) * 9] * wmsk;
#pragma unroll
      for (int tt = 0; tt < MT; ++tt) {
        v2f a;
        a.x = xp[tt][(size_t)ci * HW] * amsk[tt];
        a.y = xp[tt][(size_t)(ci + 1) * HW] * amsk[tt];
        acc[tt] = __builtin_amdgcn_wmma_f32_16x16x4_f32(
            false, a, false, bm, (short)0, acc[tt], false, false);
      }
    }
  }

  const int oc = n0 + lid;                    // D: N = lane&15
  if (oc < cout) {
    const float dm = dmod[(size_t)b * cout + oc];
    float* yb = y + ((size_t)b * cout + oc) * HW;
#pragma unroll
    for (int tt = 0; tt < MT; ++tt) {
#pragma unroll
      for (int v = 0; v < 8; ++v) {
        const int p = m0 + tt * 16 + v + half * 8;  // D: M = v + 8*(lane>=16)
        if (p < HW) {
          float val = acc[tt][v] * dm;
          yb[p] = val >= 0.f ? val : LRELU * val;
        }
      }
    }
  }
}

// ---------------------------------------------------------------------------
// rgb[b,c,p] = sum_i rw[c,i] * rsp[b,i] * x[b,i,p]  (+ prev)
__global__ void k_rgb(const float* __restrict__ x, const float* __restrict__ rsp,
                      const float* __restrict__ rw, const float* __restrict__ prev,
                      float* __restrict__ out, int cin, long hw) {
  long n = 16L * 3 * hw;
  long i = (long)blockIdx.x * blockDim.x + threadIdx.x;
  long stride = (long)gridDim.x * blockDim.x;
  for (; i < n; i += stride) {
    long p = i % hw;
    int c = (int)((i / hw) % 3);
    int b = (int)(i / (hw * 3));
    const float* xb = x + (size_t)b * cin * hw + p;
    const float* wc = rw + (size_t)c * cin;
    const float* sb = rsp + (size_t)b * cin;
    float s = 0.f;
    for (int ci = 0; ci < cin; ++ci) s += wc[ci] * sb[ci] * xb[(size_t)ci * hw];
    if (prev) s += prev[i];
    out[i] = s;
  }
}

// half-pixel bilinear upsample by integer scale, clamped edges
__global__ void k_upsample(const float* __restrict__ in, float* __restrict__ out,
                           int C, int H, int W, int S) {
  const int OH = H * S, OW = W * S;
  long n = 16L * C * OH * OW;
  long i = (long)blockIdx.x * blockDim.x + threadIdx.x;
  long stride = (long)gridDim.x * blockDim.x;
  const float inv = 1.0f / (float)S;
  for (; i < n; i += stride) {
    long tmp = i;
    int ow = (int)(tmp % OW); tmp /= OW;
    int oh = (int)(tmp % OH); tmp /= OH;   // tmp = b*C + c
    float fy = (oh + 0.5f) * inv - 0.5f;
    float fx = (ow + 0.5f) * inv - 0.5f;
    int y0 = (int)floorf(fy); float wy = fy - (float)y0;
    int x0 = (int)floorf(fx); float wx = fx - (float)x0;
    int y0c = iclamp(y0, 0, H - 1), y1c = iclamp(y0 + 1, 0, H - 1);
    int x0c = iclamp(x0, 0, W - 1), x1c = iclamp(x0 + 1, 0, W - 1);
    const float* ib = in + tmp * (long)H * W;
    float v00 = ib[(size_t)y0c * W + x0c], v01 = ib[(size_t)y0c * W + x1c];
    float v10 = ib[(size_t)y1c * W + x0c], v11 = ib[(size_t)y1c * W + x1c];
    out[i] = (1.f - wy) * ((1.f - wx) * v00 + wx * v01) +
             wy * ((1.f - wx) * v10 + wx * v11);
  }
}

// 3x3 binomial blur /16, reflect padding (C=3)
__global__ void k_blur(const float* __restrict__ in, float* __restrict__ out,
                       int H, int W) {
  long n = 16L * 3 * H * W;
  long i = (long)blockIdx.x * blockDim.x + threadIdx.x;
  long stride = (long)gridDim.x * blockDim.x;
  const float fw[3] = {1.f, 2.f, 1.f};
  for (; i < n; i += stride) {
    int w0 = (int)(i % W);
    int h0 = (int)((i / W) % H);
    long bc = i / ((long)H * W);
    const float* ib = in + bc * (long)H * W;
    float acc = 0.f;
#pragma unroll
    for (int dy = -1; dy <= 1; ++dy) {
      int hh = h0 + dy;
      if (hh < 0) hh = -hh;
      if (hh >= H) hh = 2 * H - 2 - hh;
#pragma unroll
      for (int dx = -1; dx <= 1; ++dx) {
        int ww = w0 + dx;
        if (ww < 0) ww = -ww;
        if (ww >= W) ww = 2 * W - 2 - ww;
        acc += fw[dy + 1] * fw[dx + 1] * ib[(size_t)hh * W + ww];
      }
    }
    out[i] = acc * (1.0f / 16.0f);
  }
}

// ---------------------------------------------------------------------------
static inline int nblk(long n) {
  long b = (n + 255) / 256;
  return (int)(b > 1048576 ? 1048576 : b);
}

extern "C" void kernel_launch(void* const* d_in, const int* in_sizes, int n_in,
                              void* d_out, int out_size, void* d_ws, size_t ws_size,
                              hipStream_t stream) {
  (void)in_sizes; (void)n_in; (void)out_size; (void)ws_size;
  const float* styles = (const float*)d_in[0];

  // workspace layout (floats)
  float* ws = (float*)d_ws;
  size_t off = 0;
  auto alloc = [&](size_t nf) { float* p = ws + off; off += nf; return p; };
  float* BUFA = alloc(48771072);   // 16*16*504*378 max upsampled features
  float* BUFB = alloc(48771072);
  float* BUFT = alloc(24385536);   // conv1 output scratch
  float* RGB[3] = { alloc(9144576), alloc(9144576), alloc(9144576) };
  float* S1  = alloc(8192);
  float* S2  = alloc(8192);
  float* RS  = alloc(8192);
  float* D1  = alloc(8192);
  float* D2  = alloc(8192);
  float* WSQ = alloc(262144);      // max cout*cin = 256*512

  static const int CFG[6][4] = {
      {512, 256, 2, 0}, {256, 128, 7, 1}, {128, 64, 3, 1},
      {64, 32, 3, 1},   {32, 16, 2, 1},   {16, 8, 3, 0}};

  int H = 4, Wd = 3;
  float* cur = BUFA;
  float* oth = BUFB;

  // x = ones(16, 512, 4, 3)
  k_fill<<<nblk(16L * 512 * H * Wd), 256, 0, stream>>>(cur, 16L * 512 * H * Wd, 1.0f);

  int prevIdx = -1;
  for (int j = 0; j < 6; ++j) {
    const int cin = CFG[j][0], cout = CFG[j][1], scale = CFG[j][2];
    const bool up = CFG[j][3] != 0;
    void* const* P = d_in + 1 + 9 * j;
    const float* ts1_w = (const float*)P[0];
    const float* ts1_b = (const float*)P[1];
    const float* c1_w  = (const float*)P[2];
    const float* ts2_w = (const float*)P[3];
    const float* ts2_b = (const float*)P[4];
    const float* c2_w  = (const float*)P[5];
    const float* rts_w = (const float*)P[6];
    const float* rts_b = (const float*)P[7];
    const float* rgb_w = (const float*)P[8];
    const long HW = (long)H * Wd;

    // styles (+1 folded in)
    k_style_gemm<<<(cin + 15) / 16, 32, 0, stream>>>(styles, ts1_w, ts1_b, S1, cin);
    k_style_gemm<<<(cout + 15) / 16, 32, 0, stream>>>(styles, ts2_w, ts2_b, S2, cout);
    k_style_gemm<<<(cout + 15) / 16, 32, 0, stream>>>(styles, rts_w, rts_b, RS, cout);

    // conv1: modulate in place, demod in epilogue
    k_wsq<<<nblk((long)cout * cin), 256, 0, stream>>>(c1_w, WSQ, (long)cout * cin);
    k_demod<<<(16 * cout + 255) / 256, 256, 0, stream>>>(S1, WSQ, D1, cin, cout);
    k_modulate<<<nblk(16L * cin * HW), 256, 0, stream>>>(cur, S1, cin, HW);
    {
      dim3 g((unsigned)((HW + 63) / 64), (unsigned)((cout + 15) / 16), 16);
      k_conv3x3<<<g, 32, 0, stream>>>(cur, c1_w, D1, BUFT, cin, cout, H, Wd);
    }

    // conv2
    k_wsq<<<nblk((long)cout * cout), 256, 0, stream>>>(c2_w, WSQ, (long)cout * cout);
    k_demod<<<(16 * cout + 255) / 256, 256, 0, stream>>>(S2, WSQ, D2, cout, cout);
    k_modulate<<<nblk(16L * cout * HW), 256, 0, stream>>>(BUFT, S2, cout, HW);
    {
      dim3 g((unsigned)((HW + 63) / 64), (unsigned)((cout + 15) / 16), 16);
      k_conv3x3<<<g, 32, 0, stream>>>(BUFT, c2_w, D2, cur, cout, cout, H, Wd);
    }

    // to-RGB (+ prev accumulate); last block writes straight to d_out
    const int newIdx = (prevIdx + 1) % 3;
    const float* prev = prevIdx >= 0 ? RGB[prevIdx] : nullptr;
    float* rout = (j == 5) ? (float*)d_out : RGB[newIdx];
    k_rgb<<<nblk(16L * 3 * HW), 256, 0, stream>>>(cur, RS, rgb_w, prev, rout, cout, HW);

    if (up) {
      const int upsIdx = (prevIdx + 2) % 3;
      const long OHW = HW * scale * scale;
      k_upsample<<<nblk(16L * 3 * OHW), 256, 0, stream>>>(RGB[newIdx], RGB[upsIdx], 3, H, Wd, scale);
      k_blur<<<nblk(16L * 3 * OHW), 256, 0, stream>>>(RGB[upsIdx], RGB[newIdx], H * scale, Wd * scale);
      k_upsample<<<nblk(16L * cout * OHW), 256, 0, stream>>>(cur, oth, cout, H, Wd, scale);
      float* tmp = cur; cur = oth; oth = tmp;
      H *= scale; Wd *= scale;
    }
    prevIdx = newIdx;
  }
}